// Cost_Constructor_82403242541387
// MI455X (gfx1250) — compile-verified
//
#include <hip/hip_runtime.h>
#include <hip/hip_bf16.h>
#include <stdint.h>

// ---- problem constants (from the reference) ----
#define ANG    5
#define UV     25
#define NDISP  9
#define DMIN   (-4)
#define BATCH  2
#define CH     8
#define H      256
#define W      256
// ---- tiling ----
#define TILE   16          // output tile is TILE x TILE
#define HALO   8           // max |disp*(2-u)| = 4*2
#define LDIM   32          // TILE + 2*HALO
#define PLANE  (LDIM*LDIM) // 1024 floats per uv plane
#define LDSF   (UV*PLANE)  // 25600 floats = 100 KB

typedef int v4i __attribute__((ext_vector_type(4)));
typedef __attribute__((address_space(1))) v4i* gv4i_p;   // global ptr to 16B vector
typedef __attribute__((address_space(3))) v4i* lv4i_p;   // LDS ptr to 16B vector
typedef __attribute__((address_space(3))) float* as3_fp;

// Async global -> LDS, 16 bytes per active lane (CDNA5, tracked by ASYNCcnt).
__device__ __forceinline__ void async_copy_b128(const float* gp, float* lp) {
#if __has_builtin(__builtin_amdgcn_global_load_async_to_lds_b128)
    __builtin_amdgcn_global_load_async_to_lds_b128((gv4i_p)gp, (lv4i_p)lp, 0, 0);
#else
    unsigned loff = (unsigned)(unsigned long long)(as3_fp)lp;
    asm volatile("global_load_async_to_lds_b128 %0, %1, off"
                 :: "v"(loff), "v"(gp) : "memory");
#endif
}

__device__ __forceinline__ void wait_async_zero() {
#if __has_builtin(__builtin_amdgcn_s_wait_asynccnt)
    __builtin_amdgcn_s_wait_asynccnt(0);
#else
    asm volatile("s_wait_asynccnt 0" ::: "memory");
#endif
}

__global__ __launch_bounds__(256)
void cost_constructor_kernel(const float* __restrict__ fm,
                             const float* __restrict__ mask,
                             const float* __restrict__ weight,
                             float* __restrict__ out) {
    __shared__ __align__(16) float fm_lds[LDSF];

    const int tid = threadIdx.x;
    const int x0  = blockIdx.x * TILE;
    const int y0  = blockIdx.y * TILE;
    const int bc  = blockIdx.z;          // bb*CH + cc
    const int cc  = bc & (CH - 1);
    const int bb  = bc >> 3;

    // Border tiles: pre-zero the staging buffer so OOB taps read 0 (zero pad).
    const bool border = (x0 == 0) | (y0 == 0) | (x0 == W - TILE) | (y0 == H - TILE);
    if (border) {
        float4  z  = make_float4(0.f, 0.f, 0.f, 0.f);
        float4* l4 = (float4*)fm_lds;
        #pragma unroll
        for (int p = 0; p < LDSF / 4 / 256; ++p)   // 25 iters
            l4[tid + p * 256] = z;
        __syncthreads();                            // zeros committed before async writes
    }

    // ---- stage 25 planes of 32x32 (tile + halo) via async b128 ----
    const float* fm_bc = fm + (size_t)bc * UV * H * W;
    #pragma unroll
    for (int p = 0; p < LDSF / 4 / 256; ++p) {      // 25 iters, 1 float4 per thread each
        int f  = tid + p * 256;                     // float4 index == LDS offset/16
        int e  = f << 2;                            // float index
        int uv = e >> 10;
        int r  = e & (PLANE - 1);
        int ly = r >> 5;
        int lx = r & (LDIM - 1);                    // multiple of 4
        int gy = y0 - HALO + ly;
        int gx = x0 - HALO + lx;
        if ((unsigned)gy < (unsigned)H && (unsigned)gx < (unsigned)W) {
            async_copy_b128(fm_bc + (size_t)uv * (H * W) + gy * W + gx, fm_lds + e);
        }
    }

    // ---- while the DMA flies: fold weight*mask into registers ----
    const int tx = tid & (TILE - 1);
    const int ty = tid >> 4;
    const int x  = x0 + tx;
    const int y  = y0 + ty;

    float wm[UV];
    float msum = 0.f;
    const float* mp = mask + (size_t)bb * UV * H * W + (size_t)y * W + x;
    const float* wp = weight + cc * UV;
    #pragma unroll
    for (int k = 0; k < UV; ++k) {
        float mk = mp[(size_t)k * (H * W)];
        msum += mk;
        wm[k] = mk * wp[k];
    }
    const float inv_avg = 25.0f / msum;             // 1 / mask_avg

    wait_async_zero();       // this wave's async LDS writes landed
    __syncthreads();         // everyone's writes visible

    // ---- 9 disparities x 25 taps, all from LDS (all offsets constant-folded) ----
    float* outp = out + ((size_t)bc * NDISP) * (size_t)(H * W) + (size_t)y * W + x;
    const int base = (ty + HALO) * LDIM + (tx + HALO);
    #pragma unroll
    for (int di = 0; di < NDISP; ++di) {
        const int disp = di + DMIN;
        float acc = 0.f;
        int k = 0;
        #pragma unroll
        for (int u = 0; u < ANG; ++u) {
            const int rowoff = base + disp * (2 - u) * LDIM;
            #pragma unroll
            for (int v = 0; v < ANG; ++v, ++k) {
                const int off = rowoff + disp * (2 - v);
                acc += wm[k] * fm_lds[k * PLANE + off];
            }
        }
        // Output is write-once streaming data: non-temporal so it doesn't evict
        // the L2-resident feature map / mask (118 MB, fits the 192 MB L2).
        __builtin_nontemporal_store(__builtin_floorf(acc * inv_avg),
                                    outp + (size_t)di * (H * W));
    }
}

extern "C" void kernel_launch(void* const* d_in, const int* in_sizes, int n_in,
                              void* d_out, int out_size, void* d_ws, size_t ws_size,
                              hipStream_t stream) {
    const float* fm     = (const float*)d_in[0];
    const float* mask   = (const float*)d_in[1];
    const float* weight = (const float*)d_in[2];
    float* out          = (float*)d_out;

    dim3 grid(W / TILE, H / TILE, BATCH * CH);   // 16 x 16 x 16 = 4096 workgroups
    dim3 block(256);                             // 8 waves (wave32)
    hipLaunchKernelGGL(cost_constructor_kernel, grid, block, 0, stream,
                       fm, mask, weight, out);
}